// PositionalEncoding_11854109737549
// MI455X (gfx1250) — compile-verified
//
#include <hip/hip_runtime.h>

#define B_    16
#define N_    256
#define D_    512
#define TPW   8             // t-values per wave
#define WAVES 8             // waves per 256-thread block (wave32)
#define TPB   (TPW * WAVES) // 64 t-values per block

typedef float v4f __attribute__((ext_vector_type(4)));

__global__ __launch_bounds__(256) void posenc_gather_kernel(
    const int* __restrict__ x,         // (B, N) segment lengths
    const float* __restrict__ pos_enc, // (MAX_LEN, D)
    float* __restrict__ out,           // (B, T, D)
    int T)
{
    __shared__ int s_cum[N_];
    __shared__ int s_wtot[WAVES];

    const int tid  = threadIdx.x;
    const int b    = blockIdx.y;
    const int wave = tid >> 5;
    const int lane = tid & 31;

    // ---- Stage x[b, :] into LDS via CDNA5 async global->LDS copy ----
    {
        unsigned lds_off = (unsigned)(size_t)(&s_cum[tid]);  // low 32 bits of flat shared addr = LDS offset
        unsigned long long gaddr = (unsigned long long)(size_t)(x + (size_t)b * N_ + tid);
        asm volatile("global_load_async_to_lds_b32 %0, %1, off"
                     :: "v"(lds_off), "v"(gaddr)
                     : "memory");
        asm volatile("s_wait_asynccnt 0" ::: "memory");
    }
    __syncthreads();

    // ---- Inclusive scan: wave32 shuffle scan + 8-wave combine (3 barriers total) ----
    int v = s_cum[tid];
    #pragma unroll
    for (int o = 1; o < 32; o <<= 1) {
        int n = __shfl_up(v, o, 32);
        if (lane >= o) v += n;
    }
    if (lane == 31) s_wtot[wave] = v;
    __syncthreads();

    int woff = 0;
    #pragma unroll
    for (int w = 0; w < WAVES; ++w)
        woff += (w < wave) ? s_wtot[w] : 0;
    v += woff;
    s_cum[tid] = v;          // safe: all reads of s_cum happened before previous barrier
    __syncthreads();

    const int total = s_cum[N_ - 1];   // row length (cum[-1])
    const int col   = lane * 4;        // float4 column within the 512-wide row

    #pragma unroll
    for (int tt = 0; tt < TPW; ++tt) {
        const int t = blockIdx.x * TPB + wave * TPW + tt;  // uniform across the wave
        if (t >= T) continue;

        const size_t orow = ((size_t)b * T + t) * D_;
        float* dst = out + orow + col;

        if (t < total) {
            // seg = #(cum <= t): branchless binary count over sorted LDS array (8 probes)
            int seg = 0;
            #pragma unroll
            for (int step = 128; step > 0; step >>= 1) {
                if (s_cum[seg + step - 1] <= t) seg += step;
            }
            const int start = (seg > 0) ? s_cum[seg - 1] : 0;
            const int pos   = t - start;                   // < 63 -> pos_enc rows stay L2-hot

            const float* src = pos_enc + (size_t)pos * D_ + col;
            #pragma unroll
            for (int j = 0; j < 4; ++j) {
                v4f val = *(const v4f*)(src + j * 128);                  // global_load_b128 (cached, hot)
                __builtin_nontemporal_store(val, (v4f*)(dst + j * 128)); // global_store_b128 NT (streaming)
            }
        } else {
            const v4f z = {0.0f, 0.0f, 0.0f, 0.0f};
            #pragma unroll
            for (int j = 0; j < 4; ++j)
                __builtin_nontemporal_store(z, (v4f*)(dst + j * 128));
        }
    }
}

extern "C" void kernel_launch(void* const* d_in, const int* in_sizes, int n_in,
                              void* d_out, int out_size, void* d_ws, size_t ws_size,
                              hipStream_t stream) {
    const int*   x       = (const int*)d_in[0];    // (B, N) int32
    const float* pos_enc = (const float*)d_in[1];  // (MAX_LEN, D) float32
    float*       out     = (float*)d_out;          // (B, T, D) float32

    const int T = out_size / (B_ * D_);
    if (T <= 0) return;

    dim3 grid((T + TPB - 1) / TPB, B_);
    posenc_gather_kernel<<<grid, 256, 0, stream>>>(x, pos_enc, out, T);
}